// BEVDeformableAggregation_38216619000367
// MI455X (gfx1250) — compile-verified
//
#include <hip/hip_runtime.h>
#include <math.h>

// ---------------------------------------------------------------------------
// BEV deformable-DETR decoder for gfx1250 (MI455X).
//   B=8, NQ=1024, D=256, L=3, NH=8, NP=8, HD=32, FFN=1024, H=W=200.
// Dense GEMMs: V_WMMA_F32_16X16X4_F32 (full fp32; dense phase is far under
// the 23.3 TB/s memory roofline set by the BEV sampling, so no need for f16).
// Workspace requirement: ~158 MB (see allocator below).
// ---------------------------------------------------------------------------

typedef float v2f __attribute__((ext_vector_type(2)));
typedef float v8f __attribute__((ext_vector_type(8)));

#define FROWS 8192            // B*NQ
#define DD    256
#define HBEV  200
#define WBEV  200
#define HWB   40000
#define LDSP  36              // LDS row stride (floats): 16B-aligned + conflict-free

// ---------------------------------------------------------------------------
// f32 WMMA GEMM:  Y[M,N] = X[M,K] @ W[N,K]^T + bias, optional ReLU.
// Block: 256 threads = 8 waves (4x2); block tile 64x64; wave tile 16x32.
// K chunk 32.  M%64==0, N%64==0, K%32==0 at every call site.
// Tile fill uses float4 (global_load_b128 / ds_store_b128); per chunk all
// fragments are prefetched to VGPRs, then 16 v_wmma issue back-to-back.
// ---------------------------------------------------------------------------
__global__ __launch_bounds__(256) void gemm_f32_wmma(
    const float* __restrict__ X, const float* __restrict__ W,
    const float* __restrict__ bias, float* __restrict__ Y,
    int M, int N, int K, int relu)
{
    __shared__ float Xs[64][LDSP];
    __shared__ float Ws[64][LDSP];
    const int tid  = threadIdx.x;
    const int m0   = blockIdx.x * 64;
    const int n0   = blockIdx.y * 64;
    const int wave = tid >> 5, lane = tid & 31;
    const int wr   = wave >> 1, wc = wave & 1;     // 4x2 wave grid
    const int r16  = lane & 15;
    const int hi   = lane >> 4;                    // K-half held by this lane
    const int koff = hi * 2;

    // float4 tile-fill mapping: 512 float4 per tile, 2 per thread
    const int fr = (tid >> 3) & 63;                // row within tile for j=0
    const int fc = (tid & 7) * 4;                  // float4 column offset

    v8f acc0 = {}, acc1 = {};
    for (int k0 = 0; k0 < K; k0 += 32) {
        #pragma unroll
        for (int j = 0; j < 2; ++j) {
            int r = fr + j * 32;                   // rows 0..63
            float4 vx = *(const float4*)&X[(size_t)(m0 + r) * K + k0 + fc];
            float4 vw = *(const float4*)&W[(size_t)(n0 + r) * K + k0 + fc];
            *(float4*)&Xs[r][fc] = vx;
            *(float4*)&Ws[r][fc] = vw;
        }
        __syncthreads();
        v2f af[8], bf0[8], bf1[8];
        #pragma unroll
        for (int s = 0; s < 8; ++s) {
            int kk = s * 4 + koff;
            af[s]  = *(const v2f*)&Xs[wr * 16 + r16][kk];
            bf0[s] = *(const v2f*)&Ws[wc * 32 + r16][kk];
            bf1[s] = *(const v2f*)&Ws[wc * 32 + 16 + r16][kk];
        }
        #pragma unroll
        for (int s = 0; s < 8; ++s) {
            acc0 = __builtin_amdgcn_wmma_f32_16x16x4_f32(
                false, af[s], false, bf0[s], (short)0, acc0, false, false);
            acc1 = __builtin_amdgcn_wmma_f32_16x16x4_f32(
                false, af[s], false, bf1[s], (short)0, acc1, false, false);
        }
        __syncthreads();
    }
    const int col0 = n0 + wc * 32 + r16;
    const int col1 = col0 + 16;
    const float bv0 = bias ? bias[col0] : 0.f;
    const float bv1 = bias ? bias[col1] : 0.f;
    #pragma unroll
    for (int i = 0; i < 8; ++i) {                  // C/D: M = i + 8*hi
        int row = m0 + wr * 16 + i + 8 * hi;
        float v0 = acc0[i] + bv0;
        float v1 = acc1[i] + bv1;
        if (relu) { v0 = fmaxf(v0, 0.f); v1 = fmaxf(v1, 0.f); }
        Y[(size_t)row * N + col0] = v0;
        Y[(size_t)row * N + col1] = v1;
    }
}

// ---------------------------------------------------------------------------
// Init / elementwise kernels
// ---------------------------------------------------------------------------
__global__ void init_out_kernel(const float* __restrict__ baq, float* __restrict__ o)
{
    int i = blockIdx.x * 256 + threadIdx.x;        // < FROWS*256
    o[i] = baq[i & (1024 * 256 - 1)];
}
__global__ void init_ref_kernel(const float* __restrict__ rp, float* __restrict__ rf)
{
    int i = blockIdx.x * 256 + threadIdx.x;        // < FROWS*2
    rf[i] = tanhf(rp[i & (1024 * 2 - 1)]);
}

__global__ void sine_kernel(const float* __restrict__ rf, float* __restrict__ sn)
{
    int idx = blockIdx.x * 256 + threadIdx.x;      // < FROWS*256
    int bq = idx >> 8, i = idx & 255;
    float p = (i < 128) ? rf[bq * 2 + 1] : rf[bq * 2 + 0];   // y-half then x-half
    int ii = i & 127;
    int j  = ii >> 1;
    float freq = __expf(2.9957322735539909f * (float)j * (1.f / 64.f)); // 20^(j/64)
    float t = p * 6.283185307179586f / freq;
    sn[idx] = (ii & 1) ? cosf(t) : sinf(t);
}

__global__ void qpos_kernel(const float* __restrict__ ps, const float* __restrict__ qp,
                            const float* __restrict__ o, float* __restrict__ qpos,
                            float* __restrict__ tin)
{
    int i = blockIdx.x * 256 + threadIdx.x;
    float v = ps[i] * qp[i];
    qpos[i] = v;
    tin[i]  = o[i] + v;
}

__global__ void add_kernel(const float* __restrict__ a, const float* __restrict__ b,
                           float* __restrict__ y)
{
    int i = blockIdx.x * 256 + threadIdx.x;
    y[i] = a[i] + b[i];
}

// Fused residual-add + LayerNorm over 256 features; one wave per row.
__global__ __launch_bounds__(256) void addln_kernel(
    const float* __restrict__ A, const float* __restrict__ Bv,
    const float* __restrict__ g, const float* __restrict__ be, float* __restrict__ Y)
{
    const int tid = threadIdx.x;
    const int wave = tid >> 5, lane = tid & 31;
    const int row = blockIdx.x * 8 + wave;
    const size_t base = (size_t)row * DD;
    float x[8];
    float sum = 0.f;
    #pragma unroll
    for (int j = 0; j < 8; ++j) {
        int f = lane + j * 32;
        x[j] = A[base + f] + Bv[base + f];
        sum += x[j];
    }
    #pragma unroll
    for (int o = 16; o > 0; o >>= 1) sum += __shfl_xor(sum, o, 32);
    float mean = sum * (1.f / 256.f);
    float vs = 0.f;
    #pragma unroll
    for (int j = 0; j < 8; ++j) { float d = x[j] - mean; vs += d * d; }
    #pragma unroll
    for (int o = 16; o > 0; o >>= 1) vs += __shfl_xor(vs, o, 32);
    float inv = rsqrtf(vs * (1.f / 256.f) + 1e-5f);
    #pragma unroll
    for (int j = 0; j < 8; ++j) {
        int f = lane + j * 32;
        Y[base + f] = (x[j] - mean) * inv * g[f] + be[f];
    }
}

// ---------------------------------------------------------------------------
// Self attention, flash style: one thread = one query row; K/V tiles in LDS.
// grid = B*NH*4 blocks of 256 threads (4 query blocks of 256 per (b,h)).
// ---------------------------------------------------------------------------
__global__ __launch_bounds__(256) void attn_kernel(
    const float* __restrict__ Q, const float* __restrict__ Kb,
    const float* __restrict__ Vb, float* __restrict__ O)
{
    __shared__ float Ks[128][32];
    __shared__ float Vs[128][32];
    const int blk = blockIdx.x;
    const int bh = blk >> 2, qb = blk & 3;
    const int b = bh >> 3, h = bh & 7;
    const int tid = threadIdx.x;
    const int q = qb * 256 + tid;
    const size_t rowQ = (size_t)(b * 1024 + q) * DD + h * 32;
    float qv[32];
    #pragma unroll
    for (int d = 0; d < 8; ++d) {
        float4 v = *(const float4*)&Q[rowQ + d * 4];
        qv[d * 4 + 0] = v.x; qv[d * 4 + 1] = v.y;
        qv[d * 4 + 2] = v.z; qv[d * 4 + 3] = v.w;
    }
    const float scale = 0.17677669529663687f;  // 1/sqrt(32)
    float m = -INFINITY, s = 0.f;
    float acc[32];
    #pragma unroll
    for (int d = 0; d < 32; ++d) acc[d] = 0.f;
    const int fr = tid >> 3;                       // float4 fill mapping
    const int fc = (tid & 7) * 4;
    for (int t = 0; t < 8; ++t) {
        #pragma unroll
        for (int j = 0; j < 4; ++j) {
            int kr = fr + j * 32;                  // rows 0..127
            size_t gidx = (size_t)(b * 1024 + t * 128 + kr) * DD + h * 32 + fc;
            *(float4*)&Ks[kr][fc] = *(const float4*)&Kb[gidx];
            *(float4*)&Vs[kr][fc] = *(const float4*)&Vb[gidx];
        }
        __syncthreads();
        for (int j = 0; j < 128; ++j) {
            float sc = 0.f;
            #pragma unroll
            for (int d = 0; d < 32; ++d) sc += qv[d] * Ks[j][d];
            sc *= scale;
            float nm = fmaxf(m, sc);
            float corr = __expf(m - nm);
            float p = __expf(sc - nm);
            s = s * corr + p;
            #pragma unroll
            for (int d = 0; d < 32; ++d) acc[d] = acc[d] * corr + p * Vs[j][d];
            m = nm;
        }
        __syncthreads();
    }
    float inv = 1.f / s;
    #pragma unroll
    for (int d = 0; d < 32; ++d) O[rowQ + d] = acc[d] * inv;
}

// Softmax over the 8 sampling points of each (b,q,h).
__global__ void awsm_kernel(float* __restrict__ aw)
{
    int idx = blockIdx.x * 256 + threadIdx.x;      // < FROWS*8
    size_t base = (size_t)idx * 8;
    float v[8];
    float m = -INFINITY;
    #pragma unroll
    for (int p = 0; p < 8; ++p) { v[p] = aw[base + p]; m = fmaxf(m, v[p]); }
    float s = 0.f;
    #pragma unroll
    for (int p = 0; p < 8; ++p) { v[p] = __expf(v[p] - m); s += v[p]; }
    float inv = 1.f / s;
    #pragma unroll
    for (int p = 0; p < 8; ++p) aw[base + p] = v[p] * inv;
}

// ---------------------------------------------------------------------------
// Fused deformable cross-attention: value projection (32x32) + bilinear
// sampling + attention-weighted reduction.  Uses linearity:
//   bilinear(Wv*bev + bv) = Wv * bilinear(bev) + bv * sum(valid tap weights)
// One thread = one (b,h,q); grid = B*NH*4 blocks of 256.
// ---------------------------------------------------------------------------
__global__ __launch_bounds__(256) void ca_kernel(
    const float* __restrict__ bev, const float* __restrict__ ref,
    const float* __restrict__ off, const float* __restrict__ aw,
    const float* __restrict__ vpw, const float* __restrict__ vpb,
    float* __restrict__ ca)
{
    __shared__ float Wv[32 * 32];
    __shared__ float Bvp[32];
    const int tid = threadIdx.x;
    for (int i = tid; i < 1024; i += 256) Wv[i] = vpw[i];
    if (tid < 32) Bvp[tid] = vpb[tid];
    __syncthreads();

    const int blk = blockIdx.x;
    const int bh = blk >> 2, qb = blk & 3;
    const int b = bh >> 3, h = bh & 7;
    const int q = qb * 256 + tid;
    const int bq = b * 1024 + q;
    const float rx = ref[bq * 2 + 0];
    const float ry = ref[bq * 2 + 1];
    const float* bevg = bev + ((size_t)b * DD + h * 32) * HWB;

    float wacc[32];
    #pragma unroll
    for (int c = 0; c < 32; ++c) wacc[c] = 0.f;
    float bsum = 0.f;

    for (int p = 0; p < 8; ++p) {
        const float ox = off[(size_t)bq * 128 + (h * 8 + p) * 2 + 0];
        const float oy = off[(size_t)bq * 128 + (h * 8 + p) * 2 + 1];
        const float a  = aw[(size_t)bq * 64 + h * 8 + p];
        const float lx = rx + ox * (1.f / 200.f);
        const float ly = ry + oy * (1.f / 200.f);
        const float x = (lx + 1.f) * 100.f - 0.5f;  // align_corners=False
        const float y = (ly + 1.f) * 100.f - 0.5f;
        const float xf = floorf(x), yf = floorf(y);
        const float wx1 = x - xf, wy1 = y - yf;
        const int x0 = (int)xf, y0 = (int)yf;
        #pragma unroll
        for (int tap = 0; tap < 4; ++tap) {
            const int xi = x0 + (tap & 1);
            const int yi = y0 + (tap >> 1);
            const float wt = ((tap & 1) ? wx1 : 1.f - wx1) *
                             ((tap >> 1) ? wy1 : 1.f - wy1);
            if (xi >= 0 && xi < WBEV && yi >= 0 && yi < HBEV) {
                const float coef = a * wt;
                const float* ptr = bevg + yi * WBEV + xi;
                #pragma unroll
                for (int c = 0; c < 32; ++c)
                    wacc[c] += coef * ptr[(size_t)c * HWB];
                bsum += coef;
            }
        }
    }
    float* out = ca + (size_t)bq * DD + h * 32;
    for (int d = 0; d < 32; ++d) {
        float s = Bvp[d] * bsum;
        #pragma unroll
        for (int c = 0; c < 32; ++c) s += Wv[d * 32 + c] * wacc[c];
        out[d] = s;
    }
}

// ---------------------------------------------------------------------------
// Final refine: tmp = h2 @ rr_w3^T + b3 (N=2) ; ref = tanh(tmp + atanh(clip))
// ---------------------------------------------------------------------------
__global__ __launch_bounds__(256) void rrfinal_kernel(
    const float* __restrict__ H, const float* __restrict__ w3,
    const float* __restrict__ b3, float* __restrict__ ref)
{
    __shared__ float W[512];
    __shared__ float Bb[2];
    const int tid = threadIdx.x;
    for (int i = tid; i < 512; i += 256) W[i] = w3[i];
    if (tid < 2) Bb[tid] = b3[tid];
    __syncthreads();
    const int row = blockIdx.x * 256 + tid;        // < FROWS
    const float* hr = H + (size_t)row * DD;
    float a0 = 0.f, a1 = 0.f;
    for (int k = 0; k < DD; ++k) {
        float hv = hr[k];
        a0 += hv * W[k];
        a1 += hv * W[256 + k];
    }
    a0 += Bb[0]; a1 += Bb[1];
    float r0 = ref[row * 2 + 0], r1 = ref[row * 2 + 1];
    r0 = fminf(fmaxf(r0, -1.f + 1e-5f), 1.f - 1e-5f);
    r1 = fminf(fmaxf(r1, -1.f + 1e-5f), 1.f - 1e-5f);
    ref[row * 2 + 0] = tanhf(a0 + atanhf(r0));
    ref[row * 2 + 1] = tanhf(a1 + atanhf(r1));
}

__global__ void copy_kernel(const float* __restrict__ s, float* __restrict__ d)
{
    int i = blockIdx.x * 256 + threadIdx.x;
    d[i] = s[i];
}
__global__ void packref_kernel(const float* __restrict__ rf, float* __restrict__ d)
{
    int i = blockIdx.x * 256 + threadIdx.x;        // < FROWS*2
    d[i] = rf[i] * 200.f;                          // * GRID
}

// ---------------------------------------------------------------------------
extern "C" void kernel_launch(void* const* d_in, const int* in_sizes, int n_in,
                              void* d_out, int out_size, void* d_ws, size_t ws_size,
                              hipStream_t stream)
{
    (void)in_sizes; (void)n_in; (void)out_size; (void)ws_size;
    const float* bev      = (const float*)d_in[0];
    const float* ba_query = (const float*)d_in[1];
    const float* ref_prm  = (const float*)d_in[2];
    const float* ps_w1 = (const float*)d_in[3];  const float* ps_b1 = (const float*)d_in[4];
    const float* ps_w2 = (const float*)d_in[5];  const float* ps_b2 = (const float*)d_in[6];
    const float* qp_w1 = (const float*)d_in[7];  const float* qp_b1 = (const float*)d_in[8];
    const float* qp_w2 = (const float*)d_in[9];  const float* qp_b2 = (const float*)d_in[10];
    const float* sa_in_w  = (const float*)d_in[11]; const float* sa_in_b  = (const float*)d_in[12];
    const float* sa_out_w = (const float*)d_in[13]; const float* sa_out_b = (const float*)d_in[14];
    const float* off_w1 = (const float*)d_in[15]; const float* off_b1 = (const float*)d_in[16];
    const float* off_w2 = (const float*)d_in[17]; const float* off_b2 = (const float*)d_in[18];
    const float* aw_w1  = (const float*)d_in[19]; const float* aw_b1  = (const float*)d_in[20];
    const float* aw_w2  = (const float*)d_in[21]; const float* aw_b2  = (const float*)d_in[22];
    const float* vp_w   = (const float*)d_in[23]; const float* vp_b   = (const float*)d_in[24];
    const float* op_w1  = (const float*)d_in[25]; const float* op_b1  = (const float*)d_in[26];
    const float* op_w2  = (const float*)d_in[27]; const float* op_b2  = (const float*)d_in[28];
    const float* ffn_w1 = (const float*)d_in[29]; const float* ffn_b1 = (const float*)d_in[30];
    const float* ffn_w2 = (const float*)d_in[31]; const float* ffn_b2 = (const float*)d_in[32];
    const float* ln_g   = (const float*)d_in[33]; const float* ln_b   = (const float*)d_in[34];
    const float* rr_w1  = (const float*)d_in[35]; const float* rr_b1  = (const float*)d_in[36];
    const float* rr_w2  = (const float*)d_in[37]; const float* rr_b2  = (const float*)d_in[38];
    const float* rr_w3  = (const float*)d_in[39]; const float* rr_b3  = (const float*)d_in[40];

    // ---- workspace allocator (~158 MB total) ----
    char* wsp = (char*)d_ws;
    size_t pos = 0;
    auto alloc = [&](size_t nf) { float* p = (float*)(wsp + pos); pos += nf * sizeof(float); return p; };
    const size_t F = FROWS;
    float* o    = alloc(F * 256);
    float* rf   = alloc(F * 2);
    float* sn   = alloc(F * 256);
    float* hid  = alloc(F * 1024);   // hidden of every MLP (max N = 1024)
    float* b1   = alloc(F * 256);
    float* b2   = alloc(F * 256);
    float* qpos = alloc(F * 256);
    float* tin  = alloc(F * 256);
    float* Qb   = alloc(F * 256);
    float* Kb   = alloc(F * 256);
    float* Vb   = alloc(F * 256);
    float* at   = alloc(F * 256);
    float* tg   = alloc(F * 256);
    float* qd   = alloc(F * 256);
    float* ofb  = alloc(F * 128);
    float* awb  = alloc(F * 64);
    float* cab  = alloc(F * 256);
    float* t2   = alloc(F * 256);

    auto gemm = [&](const float* X, const float* W, const float* bias, float* Y,
                    int M, int N, int K, int relu) {
        dim3 g(M / 64, N / 64);
        gemm_f32_wmma<<<g, 256, 0, stream>>>(X, W, bias, Y, M, N, K, relu);
    };

    const int EB = FROWS;            // elementwise blocks for F*256 elems / 256 thr

    init_out_kernel<<<EB, 256, 0, stream>>>(ba_query, o);
    init_ref_kernel<<<F * 2 / 256, 256, 0, stream>>>(ref_prm, rf);

    for (int l = 0; l < 3; ++l) {
        const float* sa_w = sa_in_w + (size_t)l * 768 * 256;
        const float* sa_b = sa_in_b + l * 768;

        sine_kernel<<<EB, 256, 0, stream>>>(rf, sn);
        // query_pos = mlp2(output) * mlp2(sine)
        gemm(o,   ps_w1, ps_b1, hid, F, 256, 256, 1);
        gemm(hid, ps_w2, ps_b2, b1,  F, 256, 256, 0);
        gemm(sn,  qp_w1, qp_b1, hid, F, 256, 256, 1);
        gemm(hid, qp_w2, qp_b2, b2,  F, 256, 256, 0);
        qpos_kernel<<<EB, 256, 0, stream>>>(b1, b2, o, qpos, tin);
        // self-attention: q,k from tgt_in, v from output
        gemm(tin, sa_w,             sa_b,       Qb, F, 256, 256, 0);
        gemm(tin, sa_w + 256 * 256, sa_b + 256, Kb, F, 256, 256, 0);
        gemm(o,   sa_w + 512 * 256, sa_b + 512, Vb, F, 256, 256, 0);
        attn_kernel<<<256, 256, 0, stream>>>(Qb, Kb, Vb, at);
        gemm(at, sa_out_w + (size_t)l * 256 * 256, sa_out_b + l * 256, b1, F, 256, 256, 0);
        addln_kernel<<<F / 8, 256, 0, stream>>>(o, b1,
            ln_g + (size_t)(l * 3 + 0) * 256, ln_b + (size_t)(l * 3 + 0) * 256, tg);
        // deformable cross attention
        add_kernel<<<EB, 256, 0, stream>>>(tg, qpos, qd);
        gemm(qd,  off_w1 + (size_t)l * 65536,     off_b1 + l * 256, hid, F, 256, 256, 1);
        gemm(hid, off_w2 + (size_t)l * 128 * 256, off_b2 + l * 128, ofb, F, 128, 256, 0);
        gemm(qd,  aw_w1 + (size_t)l * 65536,      aw_b1 + l * 256, hid, F, 256, 256, 1);
        gemm(hid, aw_w2 + (size_t)l * 64 * 256,   aw_b2 + l * 64,  awb, F, 64, 256, 0);
        awsm_kernel<<<F * 8 / 256, 256, 0, stream>>>(awb);
        ca_kernel<<<256, 256, 0, stream>>>(bev, rf, ofb, awb,
                                           vp_w + (size_t)l * 1024, vp_b + l * 32, cab);
        gemm(cab, op_w1 + (size_t)l * 65536, op_b1 + l * 256, hid, F, 256, 256, 1);
        gemm(hid, op_w2 + (size_t)l * 65536, op_b2 + l * 256, b1,  F, 256, 256, 0);
        addln_kernel<<<F / 8, 256, 0, stream>>>(tg, b1,
            ln_g + (size_t)(l * 3 + 1) * 256, ln_b + (size_t)(l * 3 + 1) * 256, t2);
        // FFN
        gemm(t2,  ffn_w1 + (size_t)l * 1024 * 256, ffn_b1 + l * 1024, hid, F, 1024, 256, 1);
        gemm(hid, ffn_w2 + (size_t)l * 256 * 1024, ffn_b2 + l * 256,  b1,  F, 256, 1024, 0);
        addln_kernel<<<F / 8, 256, 0, stream>>>(t2, b1,
            ln_g + (size_t)(l * 3 + 2) * 256, ln_b + (size_t)(l * 3 + 2) * 256, o);
        // ref_pos refinement
        gemm(o,   rr_w1 + (size_t)l * 65536, rr_b1 + l * 256, hid, F, 256, 256, 1);
        gemm(hid, rr_w2 + (size_t)l * 65536, rr_b2 + l * 256, b2,  F, 256, 256, 1);
        rrfinal_kernel<<<F / 256, 256, 0, stream>>>(b2, rr_w3 + (size_t)l * 512,
                                                    rr_b3 + l * 2, rf);
    }

    copy_kernel<<<EB, 256, 0, stream>>>(o, (float*)d_out);
    packref_kernel<<<F * 2 / 256, 256, 0, stream>>>(rf, (float*)d_out + F * 256);
}